// STAggBlock_2207613190271
// MI455X (gfx1250) — compile-verified
//
#include <hip/hip_runtime.h>
#include <hip/hip_bf16.h>

typedef __attribute__((ext_vector_type(2))) float v2f;
typedef __attribute__((ext_vector_type(8))) float v8f;

#define NPB   1024      // nodes per batch
#define CDIM  64        // channels
#define NBATCH 32

static __device__ __forceinline__ v8f wmma4(v2f a, v2f b, v8f c) {
  // D = A(16x4,f32) * B(4x16,f32) + C(16x16,f32)
  return __builtin_amdgcn_wmma_f32_16x16x4_f32(false, a, false, b, (short)0, c,
                                               false, false);
}

// -------------------------------------------------------------------------
// out[r, :] = scale * (A[r, :] @ W) (+ bias) (+ add[r,:]) (+ out preload)
// A: [NBATCH*NPB, 64] row-major. W: [64,64] (per-batch stride wStride floats).
// 4 waves / block, each wave owns a 16-row x 64-col tile.
// Software-pipelined K loop: operands for kk+1 are fetched during the four
// WMMAs of kk so the matrix pipe is not gated on load latency.
// -------------------------------------------------------------------------
__global__ __launch_bounds__(128) void gemm64(
    const float* __restrict__ A, const float* __restrict__ W,
    const float* __restrict__ bias, const float* __restrict__ addv,
    float* __restrict__ out, float scale, int accum, int wStride) {
  const int lane = threadIdx.x & 31;
  const int wave = threadIdx.x >> 5;
  const int tile = blockIdx.x * 4 + wave;
  const int rowBase = tile * 16;
  const int batch = rowBase >> 10;  // rows/batch = 1024
  const float* Wb = W + (size_t)batch * wStride;
  const int l16 = lane & 15, hi = lane >> 4;

  v8f acc[4];
#pragma unroll
  for (int n = 0; n < 4; ++n) {
#pragma unroll
    for (int v = 0; v < 8; ++v) {
      if (accum)
        acc[n][v] = out[(size_t)(rowBase + v + 8 * hi) * CDIM + n * 16 + l16];
      else
        acc[n][v] = 0.0f;
    }
  }

  const float* Arow = &A[(size_t)(rowBase + l16) * CDIM];

  auto loadB = [&](int kk, v2f bb[4]) {
    const int kr = kk * 4 + 2 * hi;
#pragma unroll
    for (int n = 0; n < 4; ++n) {
      bb[n].x = Wb[kr * CDIM + n * 16 + l16];
      bb[n].y = Wb[(kr + 1) * CDIM + n * 16 + l16];
    }
  };

  v2f aCur = *(const v2f*)&Arow[2 * hi];
  v2f bCur[4];
  loadB(0, bCur);

#pragma unroll
  for (int kk = 0; kk < 16; ++kk) {
    v2f aNxt = aCur;
    v2f bNxt[4] = {bCur[0], bCur[1], bCur[2], bCur[3]};
    if (kk < 15) {  // prefetch next K-slice before consuming current one
      aNxt = *(const v2f*)&Arow[(kk + 1) * 4 + 2 * hi];
      loadB(kk + 1, bNxt);
    }
#pragma unroll
    for (int n = 0; n < 4; ++n) acc[n] = wmma4(aCur, bCur[n], acc[n]);
    aCur = aNxt;
#pragma unroll
    for (int n = 0; n < 4; ++n) bCur[n] = bNxt[n];
  }

#pragma unroll
  for (int n = 0; n < 4; ++n) {
#pragma unroll
    for (int v = 0; v < 8; ++v) {
      size_t idx = (size_t)(rowBase + v + 8 * hi) * CDIM + n * 16 + l16;
      float r = acc[n][v] * scale;
      if (bias) r += bias[n * 16 + l16];
      if (addv) r += addv[idx];
      out[idx] = r;
    }
  }
}

// -------------------------------------------------------------------------
// M_b = Bm_bᵀ @ Bm_b  (64x64 per batch).  One block per batch, 16 waves,
// each wave a 16x16 tile, K = 1024.  K unrolled by 16: 16 batched loads,
// then 4 WMMAs.
// -------------------------------------------------------------------------
__global__ __launch_bounds__(512) void bgram64(const float* __restrict__ Bm,
                                               float* __restrict__ M) {
  const int b = blockIdx.x;
  const float* Bb = Bm + (size_t)b * NPB * CDIM;
  const int lane = threadIdx.x & 31;
  const int wave = threadIdx.x >> 5;  // 0..15
  const int ti16 = (wave >> 2) * 16, tj16 = (wave & 3) * 16;
  const int l16 = lane & 15, hi = lane >> 4;

  v8f acc = {0.f, 0.f, 0.f, 0.f, 0.f, 0.f, 0.f, 0.f};
  for (int k0 = 0; k0 < NPB; k0 += 16) {
    v2f a[4], bb[4];
#pragma unroll
    for (int u = 0; u < 4; ++u) {
      const int kr = k0 + u * 4 + 2 * hi;
      a[u].x = Bb[(size_t)kr * CDIM + ti16 + l16];        // A[m][k] = Bm[k][m]
      a[u].y = Bb[(size_t)(kr + 1) * CDIM + ti16 + l16];
      bb[u].x = Bb[(size_t)kr * CDIM + tj16 + l16];       // B[k][n] = Bm[k][n]
      bb[u].y = Bb[(size_t)(kr + 1) * CDIM + tj16 + l16];
    }
#pragma unroll
    for (int u = 0; u < 4; ++u) acc = wmma4(a[u], bb[u], acc);
  }
  float* Mb = M + (size_t)b * CDIM * CDIM;
#pragma unroll
  for (int v = 0; v < 8; ++v)
    Mb[(size_t)(ti16 + v + 8 * hi) * CDIM + tj16 + l16] = acc[v];
}

// -------------------------------------------------------------------------
// Flash-style diffusion step:  Hout = softmax_row(relu(cbn * E Eᵀ)) @ Vin
// per batch.  One wave per 16-row i-tile; two passes (row stats, then P·V).
// -------------------------------------------------------------------------
__global__ __launch_bounds__(32) void flash_diffuse(
    const float* __restrict__ E, const float* __restrict__ Vin,
    float* __restrict__ Hout, float cbn) {
  __shared__ float Plds[16 * 18];  // even row stride: aligned b64
  const int b = blockIdx.y;
  const float* Eb = E + (size_t)b * NPB * CDIM;
  const float* Vb = Vin + (size_t)b * NPB * CDIM;
  float* Hb = Hout + (size_t)b * NPB * CDIM;
  const int lane = threadIdx.x;
  const int l16 = lane & 15, hi = lane >> 4;
  const int ibase = blockIdx.x * 16;

  // Preload this i-tile's A operand chunks (E rows ibase..ibase+15)
  v2f qa[16];
#pragma unroll
  for (int kk = 0; kk < 16; ++kk)
    qa[kk] = *(const v2f*)&Eb[(size_t)(ibase + l16) * CDIM + kk * 4 + 2 * hi];

  float m[8], l[8];
#pragma unroll
  for (int v = 0; v < 8; ++v) { m[v] = 0.0f; l[v] = 0.0f; }  // scores >= 0

  // ---- pass 1: per-row running max + sum of exp (double-buffered loads) ----
  {
    v2f bb[16];
    const float* Er0 = &Eb[(size_t)l16 * CDIM + 2 * hi];
#pragma unroll
    for (int kk = 0; kk < 16; ++kk) bb[kk] = *(const v2f*)&Er0[kk * 4];

    for (int j = 0; j < 64; ++j) {
      v2f bn[16];
      if (j < 63) {  // prefetch next j-tile's B operand during this WMMA chain
        const float* Ern = &Eb[(size_t)((j + 1) * 16 + l16) * CDIM + 2 * hi];
#pragma unroll
        for (int kk = 0; kk < 16; ++kk) bn[kk] = *(const v2f*)&Ern[kk * 4];
      }
      v8f s = {0.f, 0.f, 0.f, 0.f, 0.f, 0.f, 0.f, 0.f};
#pragma unroll
      for (int kk = 0; kk < 16; ++kk) s = wmma4(qa[kk], bb[kk], s);
#pragma unroll
      for (int v = 0; v < 8; ++v) {
        float sv = fmaxf(s[v], 0.0f) * cbn;
        float t = sv;  // 16-lane (half-wave) max-reduce over tile columns
        t = fmaxf(t, __shfl_xor(t, 1));
        t = fmaxf(t, __shfl_xor(t, 2));
        t = fmaxf(t, __shfl_xor(t, 4));
        t = fmaxf(t, __shfl_xor(t, 8));
        float mn = fmaxf(m[v], t);
        float p = __expf(sv - mn);
        float ts = p;  // 16-lane sum-reduce
        ts += __shfl_xor(ts, 1);
        ts += __shfl_xor(ts, 2);
        ts += __shfl_xor(ts, 4);
        ts += __shfl_xor(ts, 8);
        l[v] = l[v] * __expf(m[v] - mn) + ts;
        m[v] = mn;
      }
#pragma unroll
      for (int kk = 0; kk < 16; ++kk) bb[kk] = bn[kk];
    }
  }
  float rl[8];
#pragma unroll
  for (int v = 0; v < 8; ++v) rl[v] = 1.0f / l[v];

  v8f o[4];
#pragma unroll
  for (int n = 0; n < 4; ++n)
#pragma unroll
    for (int v = 0; v < 8; ++v) o[n][v] = 0.0f;

  // ---- pass 2: recompute scores, P = exp(s-m)/l, accumulate O += P @ V ----
  for (int j = 0; j < 64; ++j) {
    const int jbase = j * 16;
    // Issue ALL loads for this tile up front: E operands feed the score
    // WMMAs immediately; V operands are consumed only after softmax + LDS
    // transpose, so they stay in flight through the whole middle section.
    v2f bb[16];
    const float* Erj = &Eb[(size_t)(jbase + l16) * CDIM + 2 * hi];
#pragma unroll
    for (int kk = 0; kk < 16; ++kk) bb[kk] = *(const v2f*)&Erj[kk * 4];
    v2f vv[4][4];
#pragma unroll
    for (int kk = 0; kk < 4; ++kk) {
      const int kr = kk * 4 + 2 * hi;
#pragma unroll
      for (int n = 0; n < 4; ++n) {
        vv[kk][n].x = Vb[(size_t)(jbase + kr) * CDIM + n * 16 + l16];
        vv[kk][n].y = Vb[(size_t)(jbase + kr + 1) * CDIM + n * 16 + l16];
      }
    }

    v8f s = {0.f, 0.f, 0.f, 0.f, 0.f, 0.f, 0.f, 0.f};
#pragma unroll
    for (int kk = 0; kk < 16; ++kk) s = wmma4(qa[kk], bb[kk], s);

    __syncthreads();  // WAR: previous iteration's P reads done
#pragma unroll
    for (int v = 0; v < 8; ++v) {
      float sv = fmaxf(s[v], 0.0f) * cbn;
      float p = __expf(sv - m[v]) * rl[v];
      Plds[(v + 8 * hi) * 18 + l16] = p;  // C-layout -> row-major tile
    }
    __syncthreads();
#pragma unroll
    for (int kk = 0; kk < 4; ++kk) {
      const int kr = kk * 4 + 2 * hi;
      v2f a = *(const v2f*)&Plds[l16 * 18 + kr];  // A-layout read of P
#pragma unroll
      for (int n = 0; n < 4; ++n) o[n] = wmma4(a, vv[kk][n], o[n]);
    }
  }

#pragma unroll
  for (int n = 0; n < 4; ++n)
#pragma unroll
    for (int v = 0; v < 8; ++v)
      Hb[(size_t)(ibase + v + 8 * hi) * CDIM + n * 16 + l16] = o[n][v];
}

// -------------------------------------------------------------------------
// out = relu(out) * bnscale over both branch outputs; zero the 2 log_p slots.
// -------------------------------------------------------------------------
__global__ void finalize_k(float* __restrict__ out, size_t n4, float bnscale) {
  size_t i = (size_t)blockIdx.x * blockDim.x + threadIdx.x;
  if (i < n4) {
    float4* o4 = (float4*)out;
    float4 v = o4[i];
    v.x = fmaxf(v.x, 0.0f) * bnscale;
    v.y = fmaxf(v.y, 0.0f) * bnscale;
    v.z = fmaxf(v.z, 0.0f) * bnscale;
    v.w = fmaxf(v.w, 0.0f) * bnscale;
    o4[i] = v;
  }
  if (i == 0) {  // log_p = zeros(2)
    out[n4 * 4 + 0] = 0.0f;
    out[n4 * 4 + 1] = 0.0f;
  }
}

extern "C" void kernel_launch(void* const* d_in, const int* in_sizes, int n_in,
                              void* d_out, int out_size, void* d_ws,
                              size_t ws_size, hipStream_t stream) {
  (void)in_sizes; (void)n_in; (void)out_size; (void)ws_size;
  const float* x    = (const float*)d_in[0];
  const float* sem  = (const float*)d_in[1];
  // t_W_out / t_b_out / s_W_out / s_b_out (idx 2,3,6,7) are dead in the ref.
  const float* tWin = (const float*)d_in[4];
  const float* tbin = (const float*)d_in[5];
  const float* sWin = (const float*)d_in[8];
  const float* sbin = (const float*)d_in[9];
  const float* xalW = (const float*)d_in[10];
  const float* xalb = (const float*)d_in[11];
  const float* xchW = (const float*)d_in[12];
  const float* xchb = (const float*)d_in[13];
  const float* salW = (const float*)d_in[14];
  const float* salb = (const float*)d_in[15];
  const float* schW = (const float*)d_in[16];
  const float* schb = (const float*)d_in[17];

  const size_t NB = (size_t)NBATCH * NPB * CDIM;  // 2,097,152 floats
  float* ws  = (float*)d_ws;
  float* Bt  = ws;            // x @ t_W_in   -> adjacency for SEM branch
  float* Bs  = Bt + NB;       // sem @ s_W_in -> adjacency for X branch
  float* Et  = Bs + NB;       // Ei3 (from x)
  float* Es  = Et + NB;       // Ei3 (from sem)
  float* h0x = Es + NB;
  float* h0s = h0x + NB;
  float* Mt  = h0s + NB;                       // [32,64,64]
  float* Ms  = Mt + (size_t)NBATCH * CDIM * CDIM;
  float* hA  = Bt;  // Bm buffers are dead once Ei3 is computed -> recycle
  float* hB  = Bs;
  float* outx = (float*)d_out;
  float* outs = outx + NB;

  const float cbn = 0.9999950000374997f;  // 1/sqrt(1 + 1e-5)
  const dim3 gG(512), bG(128);            // 2048 row-tiles, 4 waves/block
  const dim3 gF(64, 32), bF(32);          // 64 i-tiles x 32 batches

  // linears
  gemm64<<<gG, bG, 0, stream>>>(x,   tWin, tbin, nullptr, Bt,  1.0f, 0, 0);
  gemm64<<<gG, bG, 0, stream>>>(sem, sWin, sbin, nullptr, Bs,  1.0f, 0, 0);
  gemm64<<<gG, bG, 0, stream>>>(x,   xalW, xalb, nullptr, h0x, 1.0f, 0, 0);
  gemm64<<<gG, bG, 0, stream>>>(sem, salW, salb, nullptr, h0s, 1.0f, 0, 0);
  // M = BmᵀBm ; Ei3 = Bm @ M / sqrt(64)
  bgram64<<<NBATCH, 512, 0, stream>>>(Bt, Mt);
  bgram64<<<NBATCH, 512, 0, stream>>>(Bs, Ms);
  gemm64<<<gG, bG, 0, stream>>>(Bt, Mt, nullptr, nullptr, Et, 0.125f, 0, CDIM * CDIM);
  gemm64<<<gG, bG, 0, stream>>>(Bs, Ms, nullptr, nullptr, Es, 0.125f, 0, CDIM * CDIM);
  // Cheb k=0 + residual + bias (initializes d_out regions)
  gemm64<<<gG, bG, 0, stream>>>(h0x, xchW, xchb, h0x, outx, 1.0f, 0, 0);
  gemm64<<<gG, bG, 0, stream>>>(h0s, schW, schb, h0s, outs, 1.0f, 0, 0);
  // X branch: adjacency from sem (Es); 3 diffusion steps + Cheb accumulate
  flash_diffuse<<<gF, bF, 0, stream>>>(Es, h0x, hA, cbn);
  gemm64<<<gG, bG, 0, stream>>>(hA, xchW + 1 * CDIM * CDIM, nullptr, nullptr, outx, 1.0f, 1, 0);
  flash_diffuse<<<gF, bF, 0, stream>>>(Es, hA, hB, cbn);
  gemm64<<<gG, bG, 0, stream>>>(hB, xchW + 2 * CDIM * CDIM, nullptr, nullptr, outx, 1.0f, 1, 0);
  flash_diffuse<<<gF, bF, 0, stream>>>(Es, hB, hA, cbn);
  gemm64<<<gG, bG, 0, stream>>>(hA, xchW + 3 * CDIM * CDIM, nullptr, nullptr, outx, 1.0f, 1, 0);
  // SEM branch: adjacency from x (Et)
  flash_diffuse<<<gF, bF, 0, stream>>>(Et, h0s, hA, cbn);
  gemm64<<<gG, bG, 0, stream>>>(hA, schW + 1 * CDIM * CDIM, nullptr, nullptr, outs, 1.0f, 1, 0);
  flash_diffuse<<<gF, bF, 0, stream>>>(Et, hA, hB, cbn);
  gemm64<<<gG, bG, 0, stream>>>(hB, schW + 2 * CDIM * CDIM, nullptr, nullptr, outs, 1.0f, 1, 0);
  flash_diffuse<<<gF, bF, 0, stream>>>(Et, hB, hA, cbn);
  gemm64<<<gG, bG, 0, stream>>>(hA, schW + 3 * CDIM * CDIM, nullptr, nullptr, outs, 1.0f, 1, 0);
  // relu * 1/sqrt(1+eps) on both outputs, zero log_p
  const size_t ntot4 = (2 * NB) / 4;
  finalize_k<<<(unsigned)((ntot4 + 255) / 256), 256, 0, stream>>>(
      (float*)d_out, ntot4, cbn);
}